// Graclus_68169720922642
// MI455X (gfx1250) — compile-verified
//
#include <hip/hip_runtime.h>

#define N_NODES   100000
#define N_EDGES   1600000
#define HID       128
#define N_GRAPHS  128
#define N_LAYERS  4
#define N_CLASSES 10

typedef __attribute__((ext_vector_type(2))) float v2f;
typedef __attribute__((ext_vector_type(8))) float v8f;

__global__ void zero_kernel(float4* __restrict__ p, long n4) {
  long i = (long)blockIdx.x * blockDim.x + threadIdx.x;
  long stride = (long)gridDim.x * blockDim.x;
  float4 z = make_float4(0.f, 0.f, 0.f, 0.f);
  for (; i < n4; i += stride) p[i] = z;
}

__global__ void counts_kernel(const int* __restrict__ batch, float* __restrict__ counts) {
  int i = blockIdx.x * blockDim.x + threadIdx.x;
  if (i < N_NODES) unsafeAtomicAdd(&counts[batch[i]], 1.0f);
}

// One edge per wave: lane loads 4 contiguous floats of h[src], atomically adds to agg[dst].
__global__ void scatter_kernel(const float* __restrict__ h,
                               const int* __restrict__ ei,
                               float* __restrict__ agg) {
  int edge = blockIdx.x * 8 + (threadIdx.x >> 5);
  if (edge >= N_EDGES) return;
  int lane = threadIdx.x & 31;
  int s = ei[edge];
  int d = ei[edge + N_EDGES];
  float4 hv = ((const float4*)(h + (long)s * HID))[lane];
  float* ad = agg + (long)d * HID + lane * 4;
  unsafeAtomicAdd(ad + 0, hv.x);
  unsafeAtomicAdd(ad + 1, hv.y);
  unsafeAtomicAdd(ad + 2, hv.z);
  unsafeAtomicAdd(ad + 3, hv.w);
}

// Out = relu(Agg @ Wrel + bias + H @ Wroot). fp32 WMMA 16x16x4. Out may alias Agg.
// grid = M/16 blocks of 256 threads (8 waves); wave w owns output columns [16w, 16w+16).
__global__ void graphconv_gemm(const float* __restrict__ Agg,
                               const float* __restrict__ H,
                               const float* __restrict__ Wrel,
                               const float* __restrict__ Wroot,
                               const float* __restrict__ bias,
                               float* __restrict__ Out) {
  __shared__ float sA[16 * HID];
  __shared__ float sH[16 * HID];
  const int wave = threadIdx.x >> 5;
  const int lane = threadIdx.x & 31;
  const int m0 = blockIdx.x * 16;
  const int n0 = wave * 16;
  const int hi = lane >> 4;     // which K-pair half of the wave
  const int l16 = lane & 15;

  // Stage the 16x128 A tiles (agg + root features) into LDS, coalesced float4.
  for (int i = threadIdx.x * 4; i < 16 * HID; i += 256 * 4) {
    *(float4*)(sA + i) = *(const float4*)(Agg + (long)m0 * HID + i);
    *(float4*)(sH + i) = *(const float4*)(H + (long)m0 * HID + i);
  }
  __syncthreads();

  v8f acc = {};
  #pragma unroll 4
  for (int k0 = 0; k0 < HID; k0 += 4) {
    const int ka = k0 + 2 * hi;   // lanes 0-15: K=k0,k0+1 ; lanes 16-31: K=k0+2,k0+3
    v2f a, b;
    // A frag: row = l16, cols ka, ka+1
    a.x = sA[l16 * HID + ka];
    a.y = sA[l16 * HID + ka + 1];
    // B frag: rows ka, ka+1 of W, column n0+l16
    b.x = Wrel[ka * HID + n0 + l16];
    b.y = Wrel[(ka + 1) * HID + n0 + l16];
    acc = __builtin_amdgcn_wmma_f32_16x16x4_f32(false, a, false, b, (short)0, acc, false, false);
    a.x = sH[l16 * HID + ka];
    a.y = sH[l16 * HID + ka + 1];
    b.x = Wroot[ka * HID + n0 + l16];
    b.y = Wroot[(ka + 1) * HID + n0 + l16];
    acc = __builtin_amdgcn_wmma_f32_16x16x4_f32(false, a, false, b, (short)0, acc, false, false);
  }

  // C/D layout: VGPR j -> M = m0 + j + 8*hi, N = n0 + l16.
  const float bn = bias[n0 + l16];
  #pragma unroll
  for (int j = 0; j < 8; ++j) {
    const int m = m0 + j + hi * 8;
    float v = acc[j] + bn;
    v = v > 0.f ? v : 0.f;
    Out[(long)m * HID + n0 + l16] = v;
  }
}

// One node per wave: atomic-add its 128 features into Z[batch[node], col0:col0+128].
__global__ void pool_kernel(const float* __restrict__ h,
                            const int* __restrict__ batch,
                            float* __restrict__ Z, int col0) {
  int node = blockIdx.x * 8 + (threadIdx.x >> 5);
  if (node >= N_NODES) return;
  int lane = threadIdx.x & 31;
  int g = batch[node];
  float4 hv = ((const float4*)(h + (long)node * HID))[lane];
  float* zp = Z + (long)g * (N_LAYERS * HID) + col0 + lane * 4;
  unsafeAtomicAdd(zp + 0, hv.x);
  unsafeAtomicAdd(zp + 1, hv.y);
  unsafeAtomicAdd(zp + 2, hv.z);
  unsafeAtomicAdd(zp + 3, hv.w);
}

__global__ void pool_div_kernel(float* __restrict__ Z, const float* __restrict__ counts) {
  int i = blockIdx.x * 256 + threadIdx.x;   // N_GRAPHS * N_LAYERS*HID = 65536
  float c = counts[i >> 9];                 // row = i / 512
  c = c > 1.f ? c : 1.f;
  Z[i] = Z[i] / c;
}

__global__ void mlp1_kernel(const float* __restrict__ Z, const float* __restrict__ Wl1,
                            const float* __restrict__ bl1, float* __restrict__ z1) {
  int g = blockIdx.x;       // graph
  int n = threadIdx.x;      // output feature
  float acc = bl1[n];
  const float* zr = Z + (long)g * (N_LAYERS * HID);
  for (int k = 0; k < N_LAYERS * HID; ++k)
    acc += zr[k] * Wl1[k * HID + n];
  z1[(long)g * HID + n] = acc > 0.f ? acc : 0.f;
}

__global__ void mlp2_kernel(const float* __restrict__ z1, const float* __restrict__ Wl2,
                            const float* __restrict__ bl2, float* __restrict__ out) {
  int i = blockIdx.x * 128 + threadIdx.x;   // 128 * 10 = 1280 outputs
  if (i >= N_GRAPHS * N_CLASSES) return;
  int g = i / N_CLASSES, n = i % N_CLASSES;
  float acc = bl2[n];
  const float* zr = z1 + (long)g * HID;
  for (int k = 0; k < HID; ++k)
    acc += zr[k] * Wl2[k * N_CLASSES + n];
  out[i] = acc;
}

extern "C" void kernel_launch(void* const* d_in, const int* in_sizes, int n_in,
                              void* d_out, int out_size, void* d_ws, size_t ws_size,
                              hipStream_t stream) {
  (void)in_sizes; (void)n_in; (void)out_size; (void)ws_size;
  const float* x       = (const float*)d_in[0];
  const int*   ei      = (const int*)d_in[1];
  const int*   batch   = (const int*)d_in[2];
  const float* W1_rel  = (const float*)d_in[3];
  const float* b1_rel  = (const float*)d_in[4];
  const float* W1_root = (const float*)d_in[5];
  const float* Wc_rel  = (const float*)d_in[6];
  const float* bc_rel  = (const float*)d_in[7];
  const float* Wc_root = (const float*)d_in[8];
  const float* Wl1     = (const float*)d_in[9];
  const float* bl1     = (const float*)d_in[10];
  const float* Wl2     = (const float*)d_in[11];
  const float* bl2     = (const float*)d_in[12];

  float* bufA   = (float*)d_ws;                        // 100000*128
  float* bufB   = bufA + (long)N_NODES * HID;          // 100000*128
  float* Z      = bufB + (long)N_NODES * HID;          // 128*512 pooled (JK concat)
  float* counts = Z + N_GRAPHS * N_LAYERS * HID;       // 128
  float* z1     = counts + N_GRAPHS;                   // 128*128

  // Zero Z + counts + z1 (contiguous: 65536 + 128 + 16384 = 82048 floats = 20512 float4)
  zero_kernel<<<81, 256, 0, stream>>>((float4*)Z, 20512);
  counts_kernel<<<(N_NODES + 255) / 256, 256, 0, stream>>>(batch, counts);

  const float* hprev = x;
  for (int L = 0; L < N_LAYERS; ++L) {
    float* agg = (L & 1) ? bufB : bufA;
    zero_kernel<<<4096, 256, 0, stream>>>((float4*)agg, (long)N_NODES * HID / 4);
    scatter_kernel<<<(N_EDGES + 7) / 8, 256, 0, stream>>>(hprev, ei, agg);
    const float* Wr = (L == 0) ? W1_rel  : (Wc_rel  + (long)(L - 1) * HID * HID);
    const float* br = (L == 0) ? b1_rel  : (bc_rel  + (long)(L - 1) * HID);
    const float* Wo = (L == 0) ? W1_root : (Wc_root + (long)(L - 1) * HID * HID);
    graphconv_gemm<<<N_NODES / 16, 256, 0, stream>>>(agg, hprev, Wr, Wo, br, agg);
    pool_kernel<<<(N_NODES + 7) / 8, 256, 0, stream>>>(agg, batch, Z, L * HID);
    hprev = agg;
  }
  pool_div_kernel<<<(N_GRAPHS * N_LAYERS * HID) / 256, 256, 0, stream>>>(Z, counts);
  mlp1_kernel<<<N_GRAPHS, HID, 0, stream>>>(Z, Wl1, bl1, z1);
  mlp2_kernel<<<10, 128, 0, stream>>>(z1, Wl2, bl2, (float*)d_out);
}